// GraphClassifier_67095979098699
// MI455X (gfx1250) — compile-verified
//
#include <hip/hip_runtime.h>
#include <hip/hip_bf16.h>

#define HF    64      // hidden features
#define INF   128     // input features
#define OUTC  10      // classes
#define NGR   128     // graphs
#define EPSV  1e-5f

typedef float v2f __attribute__((ext_vector_type(2)));
typedef float v8f __attribute__((ext_vector_type(8)));

// ---------------------------------------------------------------------------
// CDNA5 async global->LDS copy (B128, per-lane, tracked by ASYNCcnt).
// VDST = VGPR with LDS byte offset, VADDR = VGPR pair with 64-bit address.
// ---------------------------------------------------------------------------
__device__ __forceinline__ void async_ld_b128(unsigned lds_byte_off, const float* gaddr) {
    asm volatile("global_load_async_to_lds_b128 %0, %1, off"
                 :: "v"(lds_byte_off), "v"(gaddr)
                 : "memory");
}
__device__ __forceinline__ void wait_async0() {
    asm volatile("s_wait_asynccnt 0" ::: "memory");
}

// ---------------------------------------------------------------------------
// Degree: deg[v] = 1 (self loop) + #incoming edges
// ---------------------------------------------------------------------------
__global__ void k_init_deg(float* __restrict__ deg, int n) {
    int i = blockIdx.x * blockDim.x + threadIdx.x;
    if (i < n) deg[i] = 1.0f;
}

__global__ void k_count_deg(const int* __restrict__ dst, float* __restrict__ deg, int e) {
    int i = blockIdx.x * blockDim.x + threadIdx.x;
    if (i < e) atomicAdd(&deg[dst[i]], 1.0f);
}

// ---------------------------------------------------------------------------
// t[n,64] = X[n,K] @ W[64,K]^T + b   via V_WMMA_F32_16X16X4_F32
// block = 256 threads = 8 waves; block computes a 32-row x 64-col chunk.
// X tile + W staged in LDS via GLOBAL_LOAD_ASYNC_TO_LDS_B128 (no VGPR
// round-trip), padded leading dim (K+4) so fragment reads hit distinct banks.
// ---------------------------------------------------------------------------
__global__ void k_gemm_wmma(const float* __restrict__ X,
                            const float* __restrict__ W,
                            const float* __restrict__ b,
                            float* __restrict__ Out,
                            int n, int K) {
    __shared__ float Xs[32 * (INF + 4)];   // max K = 128
    __shared__ float Ws[64 * (INF + 4)];
    const int ld = K + 4;
    const int baseRow = blockIdx.x * 32;

    // async staging: 16B chunks; row starts and padded stride are 16B-aligned
    const unsigned xs_base = (unsigned)(unsigned long long)(uintptr_t)&Xs[0];
    const unsigned ws_base = (unsigned)(unsigned long long)(uintptr_t)&Ws[0];
    const int kc = K >> 2;                 // float4 chunks per row
    for (int c = threadIdx.x; c < 32 * kc; c += 256) {
        int r = c / kc, q = c - r * kc;
        int gr = baseRow + r;
        if (gr < n)
            async_ld_b128(xs_base + (unsigned)(r * ld + q * 4) * 4u,
                          X + (long long)gr * K + q * 4);
    }
    for (int c = threadIdx.x; c < 64 * kc; c += 256) {
        int r = c / kc, q = c - r * kc;
        async_ld_b128(ws_base + (unsigned)(r * ld + q * 4) * 4u,
                      W + r * K + q * 4);  // W is [64,K] row-major
    }
    wait_async0();          // this wave's tiles are in LDS
    __syncthreads();        // everyone's tiles are in LDS

    const int wave = threadIdx.x >> 5;
    const int lane = threadIdx.x & 31;
    const int tm = wave >> 2;              // 0..1
    const int tn = wave & 3;               // 0..3
    const int l15 = lane & 15;
    const int hi  = lane >> 4;             // 0 or 1

    // C/D layout: vgpr r -> row = r + 8*hi, col = l15 (within tile)
    const int col = tn * 16 + l15;
    const float bv = b[col];
    v8f acc = { bv, bv, bv, bv, bv, bv, bv, bv };

    const int m0 = tm * 16 + l15;          // A: lane holds row m0, K pair at kb
    const int n0 = tn * 16 + l15;          // B: lane holds col n0, K pair at kb
    for (int kk = 0; kk < K; kk += 4) {
        int kb = kk + (hi << 1);
        v2f a, bb;
        a.x  = Xs[m0 * ld + kb];
        a.y  = Xs[m0 * ld + kb + 1];
        bb.x = Ws[n0 * ld + kb];           // B[k][n] = W[n][k]
        bb.y = Ws[n0 * ld + kb + 1];
        acc = __builtin_amdgcn_wmma_f32_16x16x4_f32(
            /*neg_a=*/false, a, /*neg_b=*/false, bb,
            /*c_mod=*/(short)0, acc, /*reuse_a=*/false, /*reuse_b=*/false);
    }

    #pragma unroll
    for (int r = 0; r < 8; ++r) {
        int row = baseRow + tm * 16 + r + (hi << 3);
        if (row < n) Out[(long long)row * HF + col] = acc[r];
    }
}

// ---------------------------------------------------------------------------
// agg[v] = t[v] / deg[v]  (self-loop term, norm = 1/deg). Also zeroes the
// 128-float BN stat buffer for the following stats pass.
// ---------------------------------------------------------------------------
__global__ void k_agg_init(const float* __restrict__ t, const float* __restrict__ deg,
                           float* __restrict__ agg, float* __restrict__ stats, int n) {
    int i = blockIdx.x * blockDim.x + threadIdx.x;
    if (i < 2 * HF) stats[i] = 0.0f;
    if (i >= n * 16) return;
    int v = i >> 4, f0 = (i & 15) * 4;
    float inv = 1.0f / deg[v];
    float4 tv = *reinterpret_cast<const float4*>(t + (long long)v * HF + f0);
    float4 o  = make_float4(tv.x * inv, tv.y * inv, tv.z * inv, tv.w * inv);
    *reinterpret_cast<float4*>(agg + (long long)v * HF + f0) = o;
}

// ---------------------------------------------------------------------------
// Edge scatter: 16 threads per edge, float4 per thread -> coalesced 256B
// gather of t[src]; per-float global f32 atomics into L2-resident agg.
// ---------------------------------------------------------------------------
__global__ void k_scatter(const int* __restrict__ src, const int* __restrict__ dst,
                          const float* __restrict__ deg, const float* __restrict__ t,
                          float* __restrict__ agg, int e) {
    long long i = (long long)blockIdx.x * blockDim.x + threadIdx.x;
    if (i >= (long long)e * 16) return;
    int ed = (int)(i >> 4);
    int f0 = ((int)i & 15) * 4;
    int s = src[ed], d = dst[ed];
    float nrm = rsqrtf(deg[s] * deg[d]);   // deg >= 1 -> finite
    float4 tv = *reinterpret_cast<const float4*>(t + (long long)s * HF + f0);
    float* ap = agg + (long long)d * HF + f0;
    atomicAdd(ap + 0, tv.x * nrm);
    atomicAdd(ap + 1, tv.y * nrm);
    atomicAdd(ap + 2, tv.z * nrm);
    atomicAdd(ap + 3, tv.w * nrm);
}

// ---------------------------------------------------------------------------
// BN stats: per-column sum and sum-of-squares. 4 row-lanes per column per
// block, LDS reduce, then 2 atomics per column per block.
// ---------------------------------------------------------------------------
__global__ void k_bn_stats(const float* __restrict__ agg, float* __restrict__ stats, int n) {
    __shared__ float sh[512];
    int col = threadIdx.x & 63;
    int rl  = threadIdx.x >> 6;            // 0..3
    float s = 0.0f, ss = 0.0f;
    for (int v = blockIdx.x * 4 + rl; v < n; v += gridDim.x * 4) {
        float x = agg[(long long)v * HF + col];
        s += x; ss += x * x;
    }
    sh[threadIdx.x]       = s;
    sh[256 + threadIdx.x] = ss;
    __syncthreads();
    if (rl == 0) {
        for (int j = 1; j < 4; ++j) {
            s  += sh[j * 64 + col];
            ss += sh[256 + j * 64 + col];
        }
        atomicAdd(&stats[col], s);
        atomicAdd(&stats[HF + col], ss);
    }
}

// ---------------------------------------------------------------------------
// h = relu(g * (x - mu) * rsqrt(var + eps) + be), biased variance
// ---------------------------------------------------------------------------
__global__ void k_bn_relu(const float* __restrict__ agg, const float* __restrict__ stats,
                          const float* __restrict__ g, const float* __restrict__ be,
                          float* __restrict__ h, int n) {
    int i = blockIdx.x * blockDim.x + threadIdx.x;
    if (i >= n * 16) return;
    int v = i >> 4, f0 = (i & 15) * 4;
    float invn = 1.0f / (float)n;
    float4 x = *reinterpret_cast<const float4*>(agg + (long long)v * HF + f0);
    float o[4];
    float xv[4] = { x.x, x.y, x.z, x.w };
    #pragma unroll
    for (int j = 0; j < 4; ++j) {
        int c = f0 + j;
        float mu  = stats[c] * invn;
        float var = stats[HF + c] * invn - mu * mu;
        float val = g[c] * (xv[j] - mu) * rsqrtf(var + EPSV) + be[c];
        o[j] = val > 0.0f ? val : 0.0f;
    }
    *reinterpret_cast<float4*>(h + (long long)v * HF + f0) =
        make_float4(o[0], o[1], o[2], o[3]);
}

// ---------------------------------------------------------------------------
// Sum pooling over batch segments
// ---------------------------------------------------------------------------
__global__ void k_zero_pool(float* __restrict__ pool) {
    int i = blockIdx.x * blockDim.x + threadIdx.x;
    if (i < NGR * HF) pool[i] = 0.0f;
}

__global__ void k_pool_add(const float* __restrict__ h, const int* __restrict__ batch,
                           float* __restrict__ pool, int n) {
    int i = blockIdx.x * blockDim.x + threadIdx.x;
    if (i >= n * 16) return;
    int v = i >> 4, f0 = (i & 15) * 4;
    int bidx = batch[v];
    float4 hv = *reinterpret_cast<const float4*>(h + (long long)v * HF + f0);
    float* pp = pool + (long long)bidx * HF + f0;
    atomicAdd(pp + 0, hv.x);
    atomicAdd(pp + 1, hv.y);
    atomicAdd(pp + 2, hv.z);
    atomicAdd(pp + 3, hv.w);
}

// ---------------------------------------------------------------------------
// Classifier head: out[g] = relu(pool[g] @ Wc1^T + bc1) @ Wc2^T + bc2
// 128 graphs x 64 hidden -> tiny; one block per graph.
// ---------------------------------------------------------------------------
__global__ void k_classifier(const float* __restrict__ pool,
                             const float* __restrict__ Wc1, const float* __restrict__ bc1,
                             const float* __restrict__ Wc2, const float* __restrict__ bc2,
                             float* __restrict__ out) {
    __shared__ float hid[HF];
    int gidx = blockIdx.x, f = threadIdx.x;
    float acc = bc1[f];
    const float* pg = pool + gidx * HF;
    #pragma unroll 8
    for (int k = 0; k < HF; ++k) acc += pg[k] * Wc1[f * HF + k];
    hid[f] = acc > 0.0f ? acc : 0.0f;
    __syncthreads();
    if (f < OUTC) {
        float o = bc2[f];
        #pragma unroll 8
        for (int k = 0; k < HF; ++k) o += hid[k] * Wc2[f * HF + k];
        out[gidx * OUTC + f] = o;
    }
}

// ---------------------------------------------------------------------------
extern "C" void kernel_launch(void* const* d_in, const int* in_sizes, int n_in,
                              void* d_out, int out_size, void* d_ws, size_t ws_size,
                              hipStream_t stream) {
    const float* x     = (const float*)d_in[0];
    const int*   ei    = (const int*)d_in[1];
    const int*   batch = (const int*)d_in[2];
    const float* Wl[3] = { (const float*)d_in[3], (const float*)d_in[7], (const float*)d_in[11] };
    const float* bl[3] = { (const float*)d_in[4], (const float*)d_in[8], (const float*)d_in[12] };
    const float* gl[3] = { (const float*)d_in[5], (const float*)d_in[9], (const float*)d_in[13] };
    const float* bel[3]= { (const float*)d_in[6], (const float*)d_in[10],(const float*)d_in[14] };
    const float* Wc1   = (const float*)d_in[15];
    const float* bc1   = (const float*)d_in[16];
    const float* Wc2   = (const float*)d_in[17];
    const float* bc2   = (const float*)d_in[18];
    float* out = (float*)d_out;

    const int N = in_sizes[0] / INF;
    const int E = in_sizes[1] / 2;
    const int* src = ei;
    const int* dst = ei + E;

    // workspace layout (floats)
    float* ws   = (float*)d_ws;
    size_t degN = ((size_t)N + 3) & ~(size_t)3;      // keep float4 alignment
    float* deg   = ws;
    float* bufA  = deg + degN;                        // N*64
    float* bufB  = bufA + (size_t)N * HF;             // N*64
    float* stats = bufB + (size_t)N * HF;             // 128
    float* pool  = stats + 2 * HF;                    // NGR*64

    const int T = 256;
    const int gN16 = (N * 16 + T - 1) / T;
    const long long e16 = (long long)E * 16;
    const int gE16 = (int)((e16 + T - 1) / T);

    k_init_deg<<<(N + T - 1) / T, T, 0, stream>>>(deg, N);
    k_count_deg<<<(E + T - 1) / T, T, 0, stream>>>(dst, deg, E);

    const float* layer_in = x;
    float* tb  = bufA;   // GEMM / BN-out buffer
    float* ab  = bufB;   // aggregation buffer
    int Ks[3] = { INF, HF, HF };
    for (int L = 0; L < 3; ++L) {
        int K = Ks[L];
        k_gemm_wmma<<<(N + 31) / 32, T, 0, stream>>>(layer_in, Wl[L], bl[L], tb, N, K);
        k_agg_init<<<gN16, T, 0, stream>>>(tb, deg, ab, stats, N);
        k_scatter<<<gE16, T, 0, stream>>>(src, dst, deg, tb, ab, E);
        k_bn_stats<<<512, T, 0, stream>>>(ab, stats, N);
        k_bn_relu<<<gN16, T, 0, stream>>>(ab, stats, gl[L], bel[L], tb, N);
        layer_in = tb;                 // h for next layer
        float* tmp = tb; tb = ab; ab = tmp;   // ping-pong so GEMM never writes its input
    }
    // after swap, layer_in points at the final h buffer

    k_zero_pool<<<(NGR * HF + T - 1) / T, T, 0, stream>>>(pool);
    k_pool_add<<<gN16, T, 0, stream>>>(layer_in, batch, pool, N);
    k_classifier<<<NGR, HF, 0, stream>>>(pool, Wc1, bc1, Wc2, bc2, out);
}